// LossFun_4672924418246
// MI455X (gfx1250) — compile-verified
//
#include <hip/hip_runtime.h>
#include <hip/hip_bf16.h>
#include <stdint.h>

// Problem constants (match reference setup_inputs).
#define BATCH   128
#define NPRIOR  8732
#define NTRUTH  32
#define NCLASS  2
#define THREADS 256

typedef __attribute__((ext_vector_type(2))) float v2f;
typedef __attribute__((ext_vector_type(8))) float v8f;
typedef __attribute__((ext_vector_type(4))) unsigned int v4u;
typedef __attribute__((ext_vector_type(4))) int v4i;
typedef __attribute__((ext_vector_type(8))) int v8i;

#if __has_builtin(__builtin_amdgcn_tensor_load_to_lds) && \
    __has_builtin(__builtin_amdgcn_s_wait_tensorcnt)
#define USE_TDM 1
#else
#define USE_TDM 0
#endif

// ---------------------------------------------------------------------------
// Wave32 sum reduction via one V_WMMA_F32_16X16X4_F32.
// A (16x4 f32): VGPR0 = {lanes 0-15: A[m,0]=x[m], lanes 16-31: A[m,2]=x[m+16]},
// VGPR1 (K=1/K=3) = 0. B = all-ones (layout-irrelevant for a constant matrix),
// C = 0  =>  D[m,n] = x[m] + x[m+16] for every n.
// Sum of the 8 D VGPRs gives S_low on lanes 0-15, S_high on lanes 16-31;
// one shfl_xor(16) completes the 32-lane sum on every lane.
// EXEC must be all 1s: only call from a fully-active wave.
// ---------------------------------------------------------------------------
__device__ __forceinline__ float wave_sum_wmma(float x) {
  v2f a; a.x = x;    a.y = 0.0f;
  v2f o; o.x = 1.0f; o.y = 1.0f;               // all-ones B matrix
  v8f c = {0.f, 0.f, 0.f, 0.f, 0.f, 0.f, 0.f, 0.f};
  v8f d = __builtin_amdgcn_wmma_f32_16x16x4_f32(
      /*neg_a=*/false, a, /*neg_b=*/false, o,
      /*c_mod=*/(short)0, c, /*reuse_a=*/false, /*reuse_b=*/false);
  float s = d[0] + d[1] + d[2] + d[3] + d[4] + d[5] + d[6] + d[7];
  s += __shfl_xor(s, 16, 32);
  return s;
}

// Block-wide sum: stage to LDS, wave 0 folds strided, WMMA finishes.
// Result valid on threads 0..31 (uniform across that wave).
__device__ __forceinline__ float block_sum(float v, float* buf, int nthreads) {
  const int tid = threadIdx.x;
  buf[tid] = v;
  __syncthreads();
  float tot = 0.0f;
  if (tid < 32) {
    float acc = 0.0f;
    for (int i = tid; i < nthreads; i += 32) acc += buf[i];
    tot = wave_sum_wmma(acc);
  }
  __syncthreads();   // make buf reusable by the next block_sum
  return tot;
}

__device__ __forceinline__ float smooth_l1(float d) {
  float ad = fabsf(d);
  return (ad < 1.0f) ? 0.5f * d * d : ad - 0.5f;
}

// ---------------------------------------------------------------------------
// Fused per-batch kernel: one block per batch, everything LDS-resident.
//  Stage 0: TDM loads the 32x5 targets tile into LDS (wave 0 issues
//           tensor_load_to_lds, s_wait_tensorcnt 0, then barrier).
//  Pass 1:  IoU matching (per-prior argmax over truths; per-truth best prior
//           via 64-bit shared atomicMax keys, ties -> smallest prior index).
//  Override: thread 0 applies the 32 forced matches sequentially (last-wins,
//           matching XLA scatter semantics for duplicate indices).
//  Pass 2:  encode + smooth-L1 (positives), stable 2-class logsumexp CE;
//           loss_c_mine row kept in LDS (35 KB) - no HBM round trip.
//  Select:  hard-negative mining as a radix SELECT over the LDS row:
//           loss_c only needs sum(top-k of loss_c_mine); values >= 0 and ties
//           contribute equal amounts, so tie ORDER is irrelevant to the sum.
//           4 rounds x 8 bits find T = k-th largest (float bits are order-
//           preserving for non-negative floats); result = sum_{v>T} v + rem*T.
// ---------------------------------------------------------------------------
__global__ void __launch_bounds__(THREADS)
k_main(const float* __restrict__ loc, const float* __restrict__ conf,
       const float* __restrict__ priors, const float* __restrict__ targets,
       float* __restrict__ b_np, float* __restrict__ b_ll,
       float* __restrict__ b_lcp, float* __restrict__ b_lcn)
{
  __shared__ float              s_tgt[NTRUTH * 5];   // raw targets tile (TDM dest)
  __shared__ float              s_tru[NTRUTH][4];
  __shared__ float              s_area[NTRUTH];
  __shared__ float              s_lab[NTRUTH];
  __shared__ unsigned long long s_key[NTRUTH];
  __shared__ unsigned char      s_cf[NPRIOR];        // target_conf per prior
  __shared__ unsigned char      s_ix[NPRIOR];        // best_truth_idx per prior
  __shared__ float              s_lcm[NPRIOR];       // loss_c_mine row (LDS!)
  __shared__ float              s_red[THREADS];
  __shared__ int                s_hist[256];
  __shared__ float              s_bcast;
  __shared__ unsigned           s_pref;
  __shared__ int                s_rem;

  const int b   = blockIdx.x;
  const int tid = threadIdx.x;
  const float* tgt_row = targets + (size_t)b * NTRUTH * 5;

  // ---- Stage 0: targets tile -> LDS via Tensor Data Mover ---------------
#if USE_TDM
  if (tid < 32) {                    // one wave issues the DMA
    const unsigned nelem = NTRUTH * 5;               // 160 x 4B = 640B tile
    const uint64_t ga = (uint64_t)(uintptr_t)tgt_row;
    const uint32_t la = (uint32_t)(uintptr_t)s_tgt;  // LDS byte offset
    // D# group 0: count=1 | lds_addr | global_addr[56:0] | type=2
    v4u g0;
    g0.x = 0x1u;
    g0.y = la;
    g0.z = (unsigned)(ga & 0xFFFFFFFFull);
    g0.w = (unsigned)((ga >> 32) & 0x01FFFFFFull) | (0x2u << 30);
    // D# group 1: data_size=2 (4B); tensor_dim0=160 (bits 79:48);
    // tensor_dim1=1; tile_dim0=160 (bits 127:112); tile_dim1/2=0 (1D tile);
    // tensor_dim0_stride=160; no padding / multicast / iteration.
    v8i g1;
    g1[0] = (int)(2u << 16);
    g1[1] = (int)((nelem & 0xFFFFu) << 16);
    g1[2] = (int)((nelem >> 16) | (1u << 16));
    g1[3] = (int)((nelem & 0xFFFFu) << 16);
    g1[4] = 0;
    g1[5] = (int)nelem;
    g1[6] = 0;
    g1[7] = 0;
    v4i gz = {0, 0, 0, 0};
#if defined(__clang_major__) && (__clang_major__ >= 23)
    v8i gz8 = {0, 0, 0, 0, 0, 0, 0, 0};
    __builtin_amdgcn_tensor_load_to_lds(g0, g1, gz, gz, gz8, 0);
#else
    __builtin_amdgcn_tensor_load_to_lds(g0, g1, gz, gz, 0);
#endif
    __builtin_amdgcn_s_wait_tensorcnt(0);
  }
#else
  for (int i = tid; i < NTRUTH * 5; i += THREADS) s_tgt[i] = tgt_row[i];
#endif
  __syncthreads();

  if (tid < NTRUTH) {
    float x1 = s_tgt[tid * 5 + 0], y1 = s_tgt[tid * 5 + 1];
    float x2 = s_tgt[tid * 5 + 2], y2 = s_tgt[tid * 5 + 3];
    s_tru[tid][0] = x1; s_tru[tid][1] = y1;
    s_tru[tid][2] = x2; s_tru[tid][3] = y2;
    s_area[tid] = (x2 - x1) * (y2 - y1);
    s_lab[tid]  = s_tgt[tid * 5 + 4];
    s_key[tid]  = 0ull;
  }
  __syncthreads();

  // ---- Pass 1: IoU matching --------------------------------------------
  for (int p = tid; p < NPRIOR; p += THREADS) {
    const float* pr = priors + (size_t)p * 4;
    __builtin_prefetch(pr + THREADS * 4, 0, 1);   // global_prefetch_b8
    float cx = pr[0], cy = pr[1], w = pr[2], h = pr[3];
    float px1 = cx - w * 0.5f, py1 = cy - h * 0.5f;
    float px2 = cx + w * 0.5f, py2 = cy + h * 0.5f;
    float pa  = (px2 - px1) * (py2 - py1);        // area from point form

    float best = -1.0f;
    int   bt   = 0;
    #pragma unroll 4
    for (int t = 0; t < NTRUTH; ++t) {
      float ltx = fmaxf(s_tru[t][0], px1);
      float lty = fmaxf(s_tru[t][1], py1);
      float rbx = fminf(s_tru[t][2], px2);
      float rby = fminf(s_tru[t][3], py2);
      float iw  = fmaxf(rbx - ltx, 0.0f);
      float ih  = fmaxf(rby - lty, 0.0f);
      float inter = iw * ih;
      float iou = inter / (s_area[t] + pa - inter);
      if (iou > best) { best = iou; bt = t; }     // strict >: first index wins
      // key: monotone in iou (iou >= 0); ~p breaks ties toward smallest p
      unsigned long long key =
          ((unsigned long long)__float_as_uint(iou) << 32) |
          (unsigned long long)(0xFFFFFFFFu - (unsigned)p);
      if (key > s_key[t]) atomicMax(&s_key[t], key);   // racy read = filter only
    }
    s_ix[p] = (unsigned char)bt;
    s_cf[p] = (best < 0.5f) ? (unsigned char)0
                            : (unsigned char)((int)s_lab[bt] + 1);
  }
  __syncthreads();

  // ---- Override: force each truth's best prior (overlap := 2.0 >= thresh)
  if (tid == 0) {
    for (int t = 0; t < NTRUTH; ++t) {
      unsigned bp = 0xFFFFFFFFu - (unsigned)(s_key[t] & 0xFFFFFFFFull);
      s_ix[bp] = (unsigned char)t;
      s_cf[bp] = (unsigned char)((int)s_lab[t] + 1);
    }
  }
  __syncthreads();

  // ---- Pass 2: losses ---------------------------------------------------
  float ll = 0.0f, lcp = 0.0f, np = 0.0f;
  for (int p = tid; p < NPRIOR; p += THREADS) {
    const int  cf  = s_cf[p];
    const int  t   = s_ix[p];
    const bool pos = cf > 0;

    const float* lp = loc  + ((size_t)b * NPRIOR + p) * 4;
    const float* cp = conf + ((size_t)b * NPRIOR + p) * NCLASS;
    __builtin_prefetch(lp + THREADS * 4, 0, 1);
    __builtin_prefetch(cp + THREADS * NCLASS, 0, 1);

    // cross-entropy with stable logsumexp (C = 2)
    float c0 = cp[0], c1 = cp[1];
    float m   = fmaxf(c0, c1);
    float lse = m + logf(expf(c0 - m) + expf(c1 - m));
    float g   = (cf >= 1) ? c1 : c0;          // take_along_axis (index clamped)
    float ce  = lse - g;
    s_lcm[p]  = pos ? 0.0f : ce;              // loss_c_mine stays in LDS

    if (pos) {
      lcp += ce;
      np  += 1.0f;
      const float* pr = priors + (size_t)p * 4;
      float cx = pr[0], cy = pr[1], pw = pr[2], ph = pr[3];
      float mx1 = s_tru[t][0], my1 = s_tru[t][1];
      float mx2 = s_tru[t][2], my2 = s_tru[t][3];
      float gcx = ((mx1 + mx2) * 0.5f - cx) / (0.1f * pw);
      float gcy = ((my1 + my2) * 0.5f - cy) / (0.1f * ph);
      float gw  = logf((mx2 - mx1) / pw) / 0.2f;
      float gh  = logf((my2 - my1) / ph) / 0.2f;
      ll += smooth_l1(lp[0] - gcx) + smooth_l1(lp[1] - gcy)
          + smooth_l1(lp[2] - gw)  + smooth_l1(lp[3] - gh);
    }
  }

  float tot;
  tot = block_sum(ll,  s_red, THREADS); if (tid == 0) b_ll[b]  = tot;
  tot = block_sum(lcp, s_red, THREADS); if (tid == 0) b_lcp[b] = tot;
  tot = block_sum(np,  s_red, THREADS);
  if (tid == 0) { b_np[b] = tot; s_bcast = tot; }
  __syncthreads();

  // ---- Hard-negative mining: radix select over the LDS row --------------
  const int npos = (int)s_bcast;
  int k = 3 * npos;
  if (k > NPRIOR - 1) k = NPRIOR - 1;           // num_neg = min(3*num_pos, P-1)
  if (k <= 0) { if (tid == 0) b_lcn[b] = 0.0f; return; }   // uniform exit

  if (tid == 0) { s_pref = 0u; s_rem = k; }

  for (int round = 0; round < 4; ++round) {
    const int shift = 24 - 8 * round;
    s_hist[tid] = 0;
    __syncthreads();                             // also publishes s_pref/s_rem
    const unsigned hmask = (round == 0) ? 0u : (0xFFFFFFFFu << (shift + 8));
    const unsigned pref  = s_pref;
    for (int p = tid; p < NPRIOR; p += THREADS) {
      unsigned u = __float_as_uint(s_lcm[p]);
      if ((u & hmask) == (pref & hmask))
        atomicAdd(&s_hist[(u >> shift) & 255], 1);
    }
    __syncthreads();
    if (tid == 0) {
      int rem = s_rem, acc = 0, i = 255;
      for (; i > 0; --i) {                       // scan high -> low bucket
        int cnt = s_hist[i];
        if (acc + cnt >= rem) break;
        acc += cnt;
      }
      s_pref = pref | ((unsigned)i << shift);
      s_rem  = rem - acc;
    }
    __syncthreads();
  }

  const unsigned tbits = s_pref;                 // bits of k-th largest value
  const int      rem   = s_rem;                  // ties at T still needed
  float sgt = 0.0f;
  for (int p = tid; p < NPRIOR; p += THREADS) {
    float v = s_lcm[p];
    if (__float_as_uint(v) > tbits) sgt += v;
  }
  tot = block_sum(sgt, s_red, THREADS);
  if (tid == 0) b_lcn[b] = tot + (float)rem * __uint_as_float(tbits);
}

// ---------------------------------------------------------------------------
// Final reduction over batches (WMMA wave reduction again).
// ---------------------------------------------------------------------------
__global__ void __launch_bounds__(BATCH)
k_final(const float* __restrict__ b_np, const float* __restrict__ b_ll,
        const float* __restrict__ b_lcp, const float* __restrict__ b_lcn,
        float* __restrict__ out)
{
  __shared__ float s_red[BATCH];
  const int tid = threadIdx.x;
  float vn = b_np[tid];
  float vl = b_ll[tid];
  float vc = b_lcp[tid] + b_lcn[tid];
  float n = block_sum(vn, s_red, BATCH);
  float l = block_sum(vl, s_red, BATCH);
  float c = block_sum(vc, s_red, BATCH);
  if (tid == 0) {
    out[0] = l / n;
    out[1] = c / n;
  }
}

// ---------------------------------------------------------------------------
extern "C" void kernel_launch(void* const* d_in, const int* in_sizes, int n_in,
                              void* d_out, int out_size, void* d_ws, size_t ws_size,
                              hipStream_t stream) {
  const float* loc     = (const float*)d_in[0];   // (B, P, 4)
  const float* conf    = (const float*)d_in[1];   // (B, P, 2)
  const float* priors  = (const float*)d_in[2];   // (P, 4)
  const float* targets = (const float*)d_in[3];   // (B, 32, 5)
  float* out = (float*)d_out;                     // (loss_l, loss_c)

  // Workspace (floats): np[B] | ll[B] | lcp[B] | lcn[B]  (2 KB total)
  float* b_np  = (float*)d_ws;
  float* b_ll  = b_np  + BATCH;
  float* b_lcp = b_ll  + BATCH;
  float* b_lcn = b_lcp + BATCH;

  k_main <<<BATCH, THREADS, 0, stream>>>(loc, conf, priors, targets,
                                         b_np, b_ll, b_lcp, b_lcn);
  k_final<<<1, BATCH, 0, stream>>>(b_np, b_ll, b_lcp, b_lcn, out);
}